// IterativeNormalization_5385888989602
// MI455X (gfx1250) — compile-verified
//
#include <hip/hip_runtime.h>

// ---------------------------------------------------------------------------
// IterNorm (ZCA whitening) for MI455X / gfx1250.
//   x:(32,48,48,512) fp32 -> N=73728 samples x C=512 channels, 4 groups of 128
// Pass 1 (k_gram):  per-group Gram Sxx = X^T X (WMMA, 4x2 wave grid) and
//                   channel sums (WMMA vs ones); x streamed to LDS by the
//                   Tensor Data Mover (double buffered, TENSORcnt pipelined).
// Pass 2 (k_ns):    cov = (Sxx - N mu mu^T)/(N-1) + eps*I, Newton-Schulz
//                   inverse-sqrt with per-wave WMMA 128^3 matmuls; also
//                   wmu = whiten @ mu for the mean-folding trick.
// Pass 3 (k_apply): out[s][c] = (X W^T)*gamma + (beta - wmu*gamma), computed
//                   as D[sample][channel] so every WMMA operand pair is a
//                   contiguous 8B load (no cross-row shuffles); TDM-streamed.
// Wave ids are computed with readfirstlane so all per-wave guards are scalar
// branches (EXEC stays all-ones around WMMA; no saveexec shadow copies).
// ---------------------------------------------------------------------------

typedef float v2f __attribute__((ext_vector_type(2)));
typedef float v8f __attribute__((ext_vector_type(8)));
typedef unsigned int u32;
typedef unsigned int uint32x4 __attribute__((ext_vector_type(4)));
typedef int int32x8 __attribute__((ext_vector_type(8)));
typedef int int32x4 __attribute__((ext_vector_type(4)));

#define C_TOT   512
#define GROUPS  4
#define MEMB    128
#define N_TOT   73728
#define CHUNK   48               // rows per LDS tile (double buffered)
#define PITCH   132              // LDS row pitch (floats) == TDM pad 128+4
#define BLKS_PER_G 96
#define ROWS_PER_BLK (N_TOT / BLKS_PER_G)            // 768
#define CHUNKS_PER_BLK (ROWS_PER_BLK / CHUNK)        // 16
#define EPSV    1e-7f

// workspace layout (float offsets)
#define OFF_SUMS 0
#define OFF_SXX  512
#define OFF_WMU  (OFF_SXX  + GROUPS * MEMB * MEMB)
#define OFF_W    (OFF_WMU  + 512)
#define OFF_P    (OFF_W    + GROUPS * MEMB * MEMB)
#define OFF_T    (OFF_P    + GROUPS * MEMB * MEMB)
#define OFF_T2   (OFF_T    + GROUPS * MEMB * MEMB)

#ifndef __has_builtin
#define __has_builtin(x) 0
#endif
#if __has_builtin(__builtin_amdgcn_tensor_load_to_lds) && defined(__AMDGCN__)
#define HAVE_TDM 1
#else
#define HAVE_TDM 0
#endif

static __device__ __forceinline__ v8f wmma_f32_16x16x4(v2f a, v2f b, v8f c) {
  // 8 args: (neg_a, A, neg_b, B, c_mod, C, reuse_a, reuse_b)
  return __builtin_amdgcn_wmma_f32_16x16x4_f32(false, a, false, b, (short)0, c,
                                               false, false);
}

// wave index as an SGPR value (uniform within the wave by construction)
static __device__ __forceinline__ int wave_id() {
  return __builtin_amdgcn_readfirstlane((int)(threadIdx.x >> 5));
}

#if HAVE_TDM
// low 32 bits of a generic LDS pointer are the LDS byte address
static __device__ __forceinline__ u32 ldsOff(const void* p) {
  return (u32)(uintptr_t)p;
}

// Issue a TDM load of a 2D fp32 tile: tile_w=128 floats x CHUNK rows, global
// row stride 512 floats, into LDS with 4-dword padding every 128 dwords
// (hardware-generated PITCH=132). Wave-level op; EXEC ignored.
static __device__ __forceinline__ void tdm_load_tile(u32 lds_byte_off,
                                                     const float* gsrc,
                                                     u32 tensor_rows) {
  const unsigned long long ga = (unsigned long long)(uintptr_t)gsrc;
  uint32x4 g0;
  g0[0] = 1u;                                          // count=1, no gather
  g0[1] = lds_byte_off;                                // lds_addr (bytes)
  g0[2] = (u32)ga;                                     // global_addr[31:0]
  g0[3] = (u32)((ga >> 32) & 0x01FFFFFFu) | (2u << 30);// addr[56:32] | type=2
  int32x8 g1;
  // data_size=4B (2<<16) | pad_enable (1<<20) | pad_interval=128dw (6<<22)
  // | pad_amount=4dw (3<<25); workgroup_mask=0
  g1[0] = (int)((2u << 16) | (1u << 20) | (6u << 22) | (3u << 25));
  g1[1] = (int)((u32)(MEMB & 0xFFFF) << 16);           // tensor_dim0 lo16
  g1[2] = (int)(((u32)MEMB >> 16) | ((tensor_rows & 0xFFFFu) << 16)); // dim0 hi | dim1 lo
  g1[3] = (int)((tensor_rows >> 16) | ((u32)(MEMB & 0xFFFF) << 16));  // dim1 hi | tile_dim0
  g1[4] = (int)(CHUNK & 0xFFFF);                       // tile_dim1; tile_dim2=0
  g1[5] = (int)C_TOT;                                  // tensor_dim0_stride lo32
  g1[6] = 0;                                           // stride hi | dim1_stride lo
  g1[7] = 0;
  const int32x4 z4 = {0, 0, 0, 0};
  const int32x8 z8 = {0, 0, 0, 0, 0, 0, 0, 0};
  // 6-arg form (clang-23 / therock headers): (g0, g1, g2, g3, g4, cpol)
  __builtin_amdgcn_tensor_load_to_lds(g0, g1, z4, z4, z8, 0);
}
#endif

// fallback cooperative staging: CHUNK rows x 128 floats, float4 coalesced
static __device__ __forceinline__ void stage_manual(float* dst, const float* src,
                                                    int tid) {
  const int quad = tid & 31;
  const int rb = tid >> 5;
  const float4* s4 = reinterpret_cast<const float4*>(src);
#pragma unroll
  for (int rr = 0; rr < CHUNK / 8; ++rr) {
    const int r = rb + rr * 8;
    float4 v = s4[(size_t)r * (C_TOT / 4) + quad];
    *reinterpret_cast<float4*>(&dst[r * PITCH + quad * 4]) = v;
  }
}

// ---------------------------------------------------------------------------
// Kernel 1: per-group Gram (X^T X over N) + channel sums (WMMA vs ones).
// Wave grid 4(M) x 2(N): each wave owns a 32x64 block -> 6 LDS pair-loads
// per 8 WMMA instead of 9.
// ---------------------------------------------------------------------------
__global__ __launch_bounds__(256) void k_gram(const float* __restrict__ x,
                                              float* __restrict__ sums,
                                              float* __restrict__ sxx) {
  __shared__ __align__(16) float Xs[2][CHUNK * PITCH];

  const int g    = blockIdx.x & (GROUPS - 1);
  const int blk  = blockIdx.x >> 2;
  const int tid  = threadIdx.x;
  const int lane = tid & 31;
  const int wave = wave_id();          // SGPR-uniform
  const int m    = lane & 15;          // M / N index within tile
  const int kk   = (lane >> 4) * 2;    // K pair base for this half-wave
  const int half = lane >> 4;
  const int mw   = wave & 3;           // M block (2 tiles = 32 rows), scalar
  const int nw   = wave >> 2;          // N block (4 tiles = 64 cols), scalar
  const int cA   = mw * 32 + m;        // A column base (channel)
  const int cB   = nw * 64 + m;        // B column base (channel)
  const int row_base = blk * ROWS_PER_BLK;

  const float* xr = x + (size_t)row_base * C_TOT + g * MEMB;

  v8f acc[2][4] = {};
  v8f accs[2] = {};
  const v2f ones = {1.f, 1.f};

  // stage chunk 0 into buffer 0
#if HAVE_TDM
  if (wave == 0) {
    tdm_load_tile(ldsOff(&Xs[0][0]), xr, (u32)(N_TOT - row_base));
    __builtin_amdgcn_s_wait_tensorcnt(0);
  }
#else
  stage_manual(&Xs[0][0], xr, tid);
#endif
  __syncthreads();

  for (int ch = 0; ch < CHUNKS_PER_BLK; ++ch) {
    if (ch + 1 < CHUNKS_PER_BLK) {
      const float* xn = xr + (size_t)(ch + 1) * CHUNK * C_TOT;
#if HAVE_TDM
      if (wave == 0)
        tdm_load_tile(ldsOff(&Xs[(ch + 1) & 1][0]), xn,
                      (u32)(N_TOT - (row_base + (ch + 1) * CHUNK)));
#else
      stage_manual(&Xs[(ch + 1) & 1][0], xn, tid);
#endif
    }
    const float* Xb = &Xs[ch & 1][0];
    for (int ks = 0; ks < CHUNK; ks += 4) {
      const int r0 = (ks + kk) * PITCH;
      const int r1 = r0 + PITCH;
      v2f a0, a1;                                  // A = X^T tiles (2 M tiles)
      a0.x = Xb[r0 + cA];      a0.y = Xb[r1 + cA];
      a1.x = Xb[r0 + cA + 16]; a1.y = Xb[r1 + cA + 16];
      if (nw == 0) {                               // scalar branch (SGPR nw)
        accs[0] = wmma_f32_16x16x4(a0, ones, accs[0]);
        accs[1] = wmma_f32_16x16x4(a1, ones, accs[1]);
      }
#pragma unroll
      for (int nt = 0; nt < 4; ++nt) {
        v2f b;                                     // B = X tile (4n x 16ch)
        b.x = Xb[r0 + cB + nt * 16];
        b.y = Xb[r1 + cB + nt * 16];
        acc[0][nt] = wmma_f32_16x16x4(a0, b, acc[0][nt]);
        acc[1][nt] = wmma_f32_16x16x4(a1, b, acc[1][nt]);
      }
    }
#if HAVE_TDM
    if (wave == 0 && ch + 1 < CHUNKS_PER_BLK)
      __builtin_amdgcn_s_wait_tensorcnt(0);
#endif
    __syncthreads();
  }

  float* S = sxx + g * (MEMB * MEMB);
#pragma unroll
  for (int mt = 0; mt < 2; ++mt)
#pragma unroll
    for (int nt = 0; nt < 4; ++nt)
#pragma unroll
      for (int j = 0; j < 8; ++j)
        atomicAdd(&S[(mw * 32 + mt * 16 + half * 8 + j) * MEMB +
                     nw * 64 + nt * 16 + m],
                  acc[mt][nt][j]);
  if (nw == 0) {                       // scalar; columns of accs identical
    if (m == 0) {
#pragma unroll
      for (int mt = 0; mt < 2; ++mt)
#pragma unroll
        for (int j = 0; j < 8; ++j)
          atomicAdd(&sums[g * MEMB + mw * 32 + mt * 16 + half * 8 + j],
                    accs[mt][j]);
    }
  }
}

// ---------------------------------------------------------------------------
// Per-wave 128x128x128 fp32 matmul D = A*B (row-major, pitch 128) via WMMA.
// ---------------------------------------------------------------------------
static __device__ void wave_mm128(const float* __restrict__ A,
                                  const float* __restrict__ B,
                                  float* __restrict__ D, int wave, int lane) {
  const int m = lane & 15;
  const int kk = (lane >> 4) * 2;
  const int half = lane >> 4;
  const int c0 = wave * 16;
  v8f acc[8] = {};
  for (int k0 = 0; k0 < MEMB; k0 += 4) {
    v2f a = *reinterpret_cast<const v2f*>(&A[(c0 + m) * MEMB + k0 + kk]);
#pragma unroll
    for (int t = 0; t < 8; ++t) {
      v2f b;
      b.x = B[(k0 + kk)     * MEMB + t * 16 + m];
      b.y = B[(k0 + kk + 1) * MEMB + t * 16 + m];
      acc[t] = wmma_f32_16x16x4(a, b, acc[t]);
    }
  }
#pragma unroll
  for (int t = 0; t < 8; ++t)
#pragma unroll
    for (int j = 0; j < 8; ++j)
      D[(c0 + half * 8 + j) * MEMB + t * 16 + m] = acc[t][j];
}

// ---------------------------------------------------------------------------
// Kernel 2: cov assembly + Newton-Schulz; also wmu = whiten @ mu.
// ---------------------------------------------------------------------------
__global__ __launch_bounds__(256) void k_ns(const float* __restrict__ sxx,
                                            const float* __restrict__ sums,
                                            float* __restrict__ wmuOut,
                                            float* __restrict__ Pm,
                                            float* __restrict__ Tm,
                                            float* __restrict__ T2m,
                                            float* __restrict__ Wm) {
  const int g = blockIdx.x;
  const int tid = threadIdx.x;
  const int lane = tid & 31;
  const int wave = wave_id();
  const float* S = sxx + g * MEMB * MEMB;
  float* P  = Pm  + g * MEMB * MEMB;
  float* T  = Tm  + g * MEMB * MEMB;
  float* T2 = T2m + g * MEMB * MEMB;
  float* W  = Wm  + g * MEMB * MEMB;   // holds sig, then whiten at the end

  __shared__ float sMean[MEMB];
  __shared__ float sTr;
  if (tid < MEMB) sMean[tid] = sums[g * MEMB + tid] * (1.f / (float)N_TOT);
  if (tid == 0) sTr = 0.f;
  __syncthreads();

  const float invNm1 = 1.f / ((float)N_TOT - 1.f);
  float trLoc = 0.f;
  for (int idx = tid; idx < MEMB * MEMB; idx += 256) {
    const int i = idx >> 7, j = idx & 127;
    float c = (S[idx] - (float)N_TOT * sMean[i] * sMean[j]) * invNm1;
    c = (1.f - EPSV) * c + ((i == j) ? EPSV : 0.f);
    T2[idx] = c;                       // cov parked in T2
    if (i == j) trLoc += c;
  }
  atomicAdd(&sTr, trLoc);
  __threadfence();
  __syncthreads();

  const float tr = sTr;
  const float invTr = 1.f / tr;
  for (int idx = tid; idx < MEMB * MEMB; idx += 256) {
    const int i = idx >> 7, j = idx & 127;
    W[idx] = T2[idx] * invTr;          // sig
    P[idx] = (i == j) ? 1.f : 0.f;
  }
  __threadfence(); __syncthreads();

  for (int it = 0; it < 3; ++it) {
    wave_mm128(P,  P, T,  wave, lane); __threadfence(); __syncthreads();
    wave_mm128(T,  P, T2, wave, lane); __threadfence(); __syncthreads();
    wave_mm128(T2, W, T,  wave, lane); __threadfence(); __syncthreads();
    for (int idx = tid; idx < MEMB * MEMB; idx += 256)
      P[idx] = 1.5f * P[idx] - 0.5f * T[idx];
    __threadfence(); __syncthreads();
  }

  const float s = rsqrtf(tr);
  for (int idx = tid; idx < MEMB * MEMB; idx += 256)
    W[idx] = P[idx] * s;               // whiten = P / sqrt(tr)
  __threadfence(); __syncthreads();

  if (tid < MEMB) {                    // wmu = whiten @ mu (mean folding)
    float acc = 0.f;
    for (int k = 0; k < MEMB; ++k) acc += W[tid * MEMB + k] * sMean[k];
    wmuOut[g * MEMB + tid] = acc;
  }
}

// ---------------------------------------------------------------------------
// Kernel 3: out[s][c] = (whiten @ x)[c][s] * gamma[c] + (beta - wmu*gamma)[c]
// computed as D[M=sample][N=channel] = X^T(tile) x W^T(tile): both operand
// pairs are contiguous 8B loads (LDS conflict-free with PITCH=132).
// ---------------------------------------------------------------------------
__global__ __launch_bounds__(256) void k_apply(const float* __restrict__ x,
                                               const float* __restrict__ wmu,
                                               const float* __restrict__ Wm,
                                               const float* __restrict__ gamma,
                                               const float* __restrict__ beta,
                                               float* __restrict__ out) {
  __shared__ __align__(16) float Xs[2][CHUNK * PITCH];
  __shared__ __align__(16) float sGamma[MEMB];
  __shared__ __align__(16) float sBetaP[MEMB];

  const int g    = blockIdx.x & (GROUPS - 1);
  const int blk  = blockIdx.x >> 2;
  const int tid  = threadIdx.x;
  const int lane = tid & 31;
  const int wave = wave_id();          // SGPR-uniform
  const int m    = lane & 15;
  const int kk   = (lane >> 4) * 2;
  const int half = lane >> 4;
  const int row_base = blk * ROWS_PER_BLK;
  const int cw   = wave * 16;          // this wave's 16-channel output strip

  if (tid < MEMB) {
    const float gm = gamma[g * MEMB + tid];
    sGamma[tid] = gm;
    sBetaP[tid] = beta[g * MEMB + tid] - wmu[g * MEMB + tid] * gm;
  }
  const float* Wg = Wm + g * MEMB * MEMB;
  const float* xr = x + (size_t)row_base * C_TOT + g * MEMB;

#if HAVE_TDM
  if (wave == 0) {
    tdm_load_tile(ldsOff(&Xs[0][0]), xr, (u32)(N_TOT - row_base));
    __builtin_amdgcn_s_wait_tensorcnt(0);
  }
#else
  stage_manual(&Xs[0][0], xr, tid);
#endif
  __syncthreads();

  const float gm = sGamma[cw + m];     // per-lane channel is fixed
  const float bp = sBetaP[cw + m];

  for (int ch = 0; ch < CHUNKS_PER_BLK; ++ch) {
    if (ch + 1 < CHUNKS_PER_BLK) {
      const float* xn = xr + (size_t)(ch + 1) * CHUNK * C_TOT;
#if HAVE_TDM
      if (wave == 0)
        tdm_load_tile(ldsOff(&Xs[(ch + 1) & 1][0]), xn,
                      (u32)(N_TOT - (row_base + (ch + 1) * CHUNK)));
#else
      stage_manual(&Xs[(ch + 1) & 1][0], xn, tid);
#endif
    }
    const float* Xb = &Xs[ch & 1][0];

    v8f acc[3] = {};                   // 3 sample tiles x 16 channels
    for (int k0 = 0; k0 < MEMB; k0 += 4) {
      // B = W^T tile: B[K=k][N=c'] = Wg[c'*128 + k]; contiguous pair
      v2f b = *reinterpret_cast<const v2f*>(&Wg[(cw + m) * MEMB + k0 + kk]);
#pragma unroll
      for (int t = 0; t < 3; ++t) {
        // A = X^T tile: A[M=sample][K=chan]; contiguous pair, conflict-free
        v2f a = *reinterpret_cast<const v2f*>(&Xb[(t * 16 + m) * PITCH + k0 + kk]);
        acc[t] = wmma_f32_16x16x4(a, b, acc[t]);
      }
    }

    const int row0 = row_base + ch * CHUNK;
#pragma unroll
    for (int t = 0; t < 3; ++t) {
#pragma unroll
      for (int j = 0; j < 8; ++j) {
        const int srow = row0 + t * 16 + j + 8 * half;
        out[(size_t)srow * C_TOT + g * MEMB + cw + m] = acc[t][j] * gm + bp;
      }
    }
#if HAVE_TDM
    if (wave == 0 && ch + 1 < CHUNKS_PER_BLK)
      __builtin_amdgcn_s_wait_tensorcnt(0);
#endif
    __syncthreads();
  }
}

// ---------------------------------------------------------------------------
extern "C" void kernel_launch(void* const* d_in, const int* in_sizes, int n_in,
                              void* d_out, int out_size, void* d_ws, size_t ws_size,
                              hipStream_t stream) {
  (void)in_sizes; (void)n_in; (void)out_size; (void)ws_size;
  const float* x     = (const float*)d_in[0];
  const float* gamma = (const float*)d_in[1];
  const float* beta  = (const float*)d_in[2];
  float* out = (float*)d_out;
  float* ws  = (float*)d_ws;

  float* sums = ws + OFF_SUMS;
  float* sxx  = ws + OFF_SXX;
  float* wmu  = ws + OFF_WMU;
  float* W    = ws + OFF_W;
  float* P    = ws + OFF_P;
  float* T    = ws + OFF_T;
  float* T2   = ws + OFF_T2;

  // zero the atomic accumulators (sums + Sxx)
  (void)hipMemsetAsync(ws, 0, (size_t)OFF_WMU * sizeof(float), stream);

  k_gram <<<dim3(GROUPS * BLKS_PER_G), dim3(256), 0, stream>>>(x, sums, sxx);
  k_ns   <<<dim3(GROUPS),              dim3(256), 0, stream>>>(sxx, sums, wmu, P, T, T2, W);
  k_apply<<<dim3(GROUPS * BLKS_PER_G), dim3(256), 0, stream>>>(x, wmu, W, gamma, beta, out);
}